// PairRepresentationBuilder_14164802142800
// MI455X (gfx1250) — compile-verified
//
#include <hip/hip_runtime.h>
#include <hip/hip_bf16.h>

typedef __attribute__((ext_vector_type(2))) float v2f;
typedef __attribute__((ext_vector_type(4))) float f4;
typedef __attribute__((ext_vector_type(8))) float v8f;

#define SDIM 256
#define PDIM 64
#define NSEQ 1024
#define NB   2

// ---------------------------------------------------------------------------
// Phase 1: fused projection GEMM with V_WMMA_F32_16X16X4_F32.
//   S  : (2048, 256) row-major   (single_repr flattened over B,N)
//   W  : (512, 64)   row-major   (rows 0..255 = W_i, rows 256..511 = W_j)
//   P  : (2048, 128) row-major   (cols 0..63 = proj_i, cols 64..127 = proj_j)
// One wave computes one 16x16 tile of P, looping K=256 in steps of 4.
// ---------------------------------------------------------------------------
__global__ __launch_bounds__(32) void pair_proj_wmma(
    const float* __restrict__ S,
    const float* __restrict__ W,
    float* __restrict__ P)
{
  const int lane = threadIdx.x;       // 0..31, wave32
  const int m0   = blockIdx.x * 16;   // row tile in [0,2048)
  const int n0   = blockIdx.y * 16;   // col tile in [0,128)
  const int half = lane >> 4;         // 0: K 0,1   1: K 2,3  (ISA 16x4 A layout)
  const int l15  = lane & 15;

  // A: lane holds row m, K = 2*half + {0,1}
  const float* srow = S + (size_t)(m0 + l15) * SDIM + 2 * half;
  // B: lane holds col n of the fused (256 x 128) weight; column base pointer
  const int n = n0 + l15;
  const float* wcol = W + (n < PDIM ? (size_t)n
                                    : (size_t)SDIM * PDIM + (size_t)(n - PDIM));

  v8f acc = {};
  #pragma unroll 4
  for (int k = 0; k < SDIM; k += 4) {
    v2f a, b;
    a.x = srow[k];
    a.y = srow[k + 1];
    const int kr = k + 2 * half;           // B rows K = kr, kr+1 for this half
    b.x = wcol[(size_t)kr * PDIM];
    b.y = wcol[(size_t)(kr + 1) * PDIM];
    // 8 args: (neg_a, A, neg_b, B, c_mod, C, reuse_a, reuse_b)
    acc = __builtin_amdgcn_wmma_f32_16x16x4_f32(
        false, a, false, b, (short)0, acc, false, false);
  }

  // D layout (32-bit 16x16): VGPR r -> M = r + 8*half, N = l15
  #pragma unroll
  for (int r = 0; r < 8; ++r) {
    P[(size_t)(m0 + r + 8 * half) * 128 + (n0 + l15)] = acc[r];
  }
}

// ---------------------------------------------------------------------------
// Phase 2: pairwise build + LayerNorm + ReLU + rel-pos embedding (store-bound).
// Block = 256 threads = 16 pair-groups of 16 lanes; each lane owns 4 channels.
// LN reduction over 64 channels = 4 xor-shuffles within a 16-lane half-wave.
// One block owns one (b,i) row and streams all 1024 j's with NT stores.
// ---------------------------------------------------------------------------
__global__ __launch_bounds__(256) void pair_build(
    const float* __restrict__ P,      // (2048,128)
    const float* __restrict__ bias,   // (64,)
    const float* __restrict__ gamma,  // (64,)
    const float* __restrict__ beta,   // (64,)
    const float* __restrict__ embed,  // (65,64)
    float* __restrict__ out)          // (2,1024,1024,64)
{
  const int row = blockIdx.x;          // b*1024 + i
  const int i   = row & (NSEQ - 1);
  const int t   = threadIdx.x;
  const int g   = t >> 4;              // pair slot 0..15 within block
  const int q   = t & 15;              // lane within 16-lane group
  const int p   = q * 4;               // channel base (float4)

  const f4 pi = *(const f4*)(P + (size_t)row * 128 + p);
  const f4 b4 = *(const f4*)(bias + p);
  const f4 g4 = *(const f4*)(gamma + p);
  const f4 e0 = *(const f4*)(beta + p);

  f4 base;                              // proj_i + bias, reused for all j
  base.x = pi.x + b4.x; base.y = pi.y + b4.y;
  base.z = pi.z + b4.z; base.w = pi.w + b4.w;

  const size_t jrow_base = (size_t)(row & ~(NSEQ - 1)) * 128;   // proj_j rows for this b
  const size_t out_base  = (size_t)row * NSEQ * PDIM;

  for (int jb = 0; jb < NSEQ; jb += 16) {
    const int j = jb + g;
    const f4 pj = *(const f4*)(P + jrow_base + (size_t)j * 128 + 64 + p);

    f4 x;
    x.x = base.x + pj.x; x.y = base.y + pj.y;
    x.z = base.z + pj.z; x.w = base.w + pj.w;

    float s  = x.x + x.y + x.z + x.w;
    float ss = x.x * x.x + x.y * x.y + x.z * x.z + x.w * x.w;
    #pragma unroll
    for (int off = 1; off < 16; off <<= 1) {
      s  += __shfl_xor(s,  off, 16);
      ss += __shfl_xor(ss, off, 16);
    }
    const float mean = s * (1.0f / 64.0f);
    const float var  = ss * (1.0f / 64.0f) - mean * mean;
    const float rstd = rsqrtf(var + 1e-5f);

    int rel = j - i;
    rel = rel < -32 ? -32 : (rel > 32 ? 32 : rel);
    rel += 32;
    const f4 e4 = *(const f4*)(embed + (size_t)rel * PDIM + p);

    f4 y;
    y.x = fmaxf((x.x - mean) * rstd * g4.x + e0.x, 0.0f) + e4.x;
    y.y = fmaxf((x.y - mean) * rstd * g4.y + e0.y, 0.0f) + e4.y;
    y.z = fmaxf((x.z - mean) * rstd * g4.z + e0.z, 0.0f) + e4.z;
    y.w = fmaxf((x.w - mean) * rstd * g4.w + e0.w, 0.0f) + e4.w;

    // 512 MB streamed output > 192 MB L2: non-temporal store
    __builtin_nontemporal_store(y, (f4*)(out + out_base + (size_t)j * PDIM + p));
  }
}

extern "C" void kernel_launch(void* const* d_in, const int* in_sizes, int n_in,
                              void* d_out, int out_size, void* d_ws, size_t ws_size,
                              hipStream_t stream) {
  const float* S     = (const float*)d_in[0];   // single_repr (2,1024,256)
  const float* W     = (const float*)d_in[1];   // (512,64)
  const float* bias  = (const float*)d_in[2];   // (64,)
  const float* gamma = (const float*)d_in[3];   // (64,)
  const float* beta  = (const float*)d_in[4];   // (64,)
  const float* embed = (const float*)d_in[5];   // (65,64)
  float* out = (float*)d_out;
  float* P   = (float*)d_ws;                    // (2048,128) = 1 MB scratch

  (void)in_sizes; (void)n_in; (void)out_size; (void)ws_size;

  dim3 g1((NB * NSEQ) / 16, 128 / 16);          // (128, 8) tiles, 1 wave each
  pair_proj_wmma<<<g1, 32, 0, stream>>>(S, W, P);

  pair_build<<<NB * NSEQ, 256, 0, stream>>>(P, bias, gamma, beta, embed, out);
}